// NRIMessagePassing_25142738550916
// MI455X (gfx1250) — compile-verified
//
#include <hip/hip_runtime.h>
#include <hip/hip_bf16.h>

typedef __attribute__((ext_vector_type(2))) float v2f;
typedef __attribute__((ext_vector_type(8))) float v8f;

#define D32 32

// ---------------------------------------------------------------------------
// Y[r][:] = X[r][:] @ W.T (+ bias) for D=32, using V_WMMA_F32_16X16X4_F32.
// One wave handles a 16-row tile and both 16-column halves (16 WMMAs/wave).
// A-fragment (16x4 f32): lane<16 -> {X[m0+l][k0], X[m0+l][k0+1]},
//                        lane>=16 -> {X[m0+l][k0+2], X[m0+l][k0+3]}.
// B-fragment (4x16 f32) for B[k][n]=W[n][k] has the same pattern on W.
// C/D (16x16 f32): vgpr i, lane l<16 -> (row m0+i,  col l)
//                          lane>=16  -> (row m0+i+8, col l-16).
// ---------------------------------------------------------------------------
__global__ __launch_bounds__(256) void linear32_wmma(
    const float* __restrict__ X, const float* __restrict__ W,
    const float* __restrict__ bias, float* __restrict__ Y, int nrows)
{
    const int lane = threadIdx.x & 31;
    const int wave = threadIdx.x >> 5;
    const int tile = blockIdx.x * 8 + wave;       // 16-row tile index
    const int m0   = tile * 16;
    if (m0 >= nrows) return;                      // wave-uniform exit (EXEC stays full)

    const int half = lane >> 4;                   // 0 or 1
    const int l16  = lane & 15;

    int row = m0 + l16;
    if (row >= nrows) row = nrows - 1;            // clamp (N%16==0 in practice)

    const float* xp  = X + (size_t)row * D32 + 2 * half;
    const float* wp0 = W + (size_t)l16        * D32 + 2 * half;   // columns 0..15
    const float* wp1 = W + (size_t)(16 + l16) * D32 + 2 * half;   // columns 16..31

    v8f c0 = {};
    v8f c1 = {};
#pragma unroll
    for (int k0 = 0; k0 < D32; k0 += 4) {
        v2f a  = *(const v2f*)(xp  + k0);
        v2f b0 = *(const v2f*)(wp0 + k0);
        v2f b1 = *(const v2f*)(wp1 + k0);
        c0 = __builtin_amdgcn_wmma_f32_16x16x4_f32(
                false, a, false, b0, (short)0, c0, false, false);
        c1 = __builtin_amdgcn_wmma_f32_16x16x4_f32(
                false, a, false, b1, (short)0, c1, false, false);
    }

    if (bias) {
        const float bv0 = bias[l16];
        const float bv1 = bias[16 + l16];
#pragma unroll
        for (int i = 0; i < 8; ++i) { c0[i] += bv0; c1[i] += bv1; }
    }

#pragma unroll
    for (int i = 0; i < 8; ++i) {
        const int r = m0 + i + 8 * half;
        if (r < nrows) {
            Y[(size_t)r * D32 + l16]      = c0[i];
            Y[(size_t)r * D32 + 16 + l16] = c1[i];
        }
    }
}

// deg[v] += 1 for each edge (u->v); deg kept as float for later scaling.
__global__ __launch_bounds__(256) void degree_kernel(
    const int* __restrict__ dst, float* __restrict__ deg, int E)
{
    const int e = blockIdx.x * blockDim.x + threadIdx.x;
    if (e < E) atomicAdd(&deg[dst[e]], 1.0f);
}

// agg[v][:] = deg[v] * dst_feat[v][:]   (float4-vectorized over N*8 threads)
__global__ __launch_bounds__(256) void init_agg_kernel(
    const float* __restrict__ deg, const float* __restrict__ dst_feat,
    float* __restrict__ agg, int n4)   // n4 = N * 8
{
    const int t = blockIdx.x * blockDim.x + threadIdx.x;
    if (t >= n4) return;
    const float  s = deg[t >> 3];
    const float4 v = ((const float4*)dst_feat)[t];
    float4 r; r.x = s * v.x; r.y = s * v.y; r.z = s * v.z; r.w = s * v.w;
    ((float4*)agg)[t] = r;
}

// agg[dst[e]][:] += src_feat[src[e]][:]  via f32 global atomics.
// One thread handles 4 consecutive floats of one edge (E*8 threads).
__global__ __launch_bounds__(256) void scatter_add_kernel(
    const int* __restrict__ src, const int* __restrict__ dst,
    const float* __restrict__ src_feat, float* __restrict__ agg, int E)
{
    const int t = blockIdx.x * blockDim.x + threadIdx.x;
    if (t >= E * 8) return;
    const int e = t >> 3;
    const int c = (t & 7) * 4;
    const int s = src[e];
    const int d = dst[e];
    const float4 v = *(const float4*)(src_feat + (size_t)s * D32 + c);
    float* p = agg + (size_t)d * D32 + c;
    atomicAdd(p + 0, v.x);
    atomicAdd(p + 1, v.y);
    atomicAdd(p + 2, v.z);
    atomicAdd(p + 3, v.w);
}

// out[e][:] = src2[src[e]][:] + dst2[dst[e]][:]   (float4 per thread)
__global__ __launch_bounds__(256) void edge_out_kernel(
    const int* __restrict__ src, const int* __restrict__ dst,
    const float* __restrict__ src2, const float* __restrict__ dst2,
    float* __restrict__ out, int E)
{
    const int t = blockIdx.x * blockDim.x + threadIdx.x;
    if (t >= E * 8) return;
    const int e = t >> 3;
    const int c = (t & 7) * 4;
    const int s = src[e];
    const int d = dst[e];
    const float4 a = *(const float4*)(src2 + (size_t)s * D32 + c);
    const float4 b = *(const float4*)(dst2 + (size_t)d * D32 + c);
    float4 r; r.x = a.x + b.x; r.y = a.y + b.y; r.z = a.z + b.z; r.w = a.w + b.w;
    *(float4*)(out + (size_t)e * D32 + c) = r;
}

extern "C" void kernel_launch(void* const* d_in, const int* in_sizes, int n_in,
                              void* d_out, int out_size, void* d_ws, size_t ws_size,
                              hipStream_t stream)
{
    // setup_inputs order: feat, src, dst, Ws, Wd, bd, Wr, br
    const float* feat = (const float*)d_in[0];
    const int*   src  = (const int*)  d_in[1];
    const int*   dst  = (const int*)  d_in[2];
    const float* Ws   = (const float*)d_in[3];
    const float* Wd   = (const float*)d_in[4];
    const float* bd   = (const float*)d_in[5];
    const float* Wr   = (const float*)d_in[6];
    const float* br   = (const float*)d_in[7];
    float*       out  = (float*)d_out;

    const int N = in_sizes[0] / D32;
    const int E = in_sizes[1];

    // Workspace layout (floats)
    float* ws       = (float*)d_ws;
    const size_t nd = (size_t)N * D32;
    float* src_feat = ws;
    float* dst_feat = src_feat + nd;
    float* agg      = dst_feat + nd;
    float* outbuf   = agg      + nd;
    float* src2     = outbuf   + nd;
    float* dst2     = src2     + nd;
    float* deg      = dst2     + nd;

    const int tiles      = (N + 15) / 16;
    const int gemm_grid  = (tiles + 7) / 8;          // 8 waves (16-row tiles) per block
    const int edge_grid  = (E * 8 + 255) / 256;      // float4 per thread
    const int node4_grid = (N * 8 + 255) / 256;
    const int e_grid     = (E + 255) / 256;

    // 1) degree count
    hipMemsetAsync(deg, 0, (size_t)N * sizeof(float), stream);
    degree_kernel<<<e_grid, 256, 0, stream>>>(dst, deg, E);

    // 2) src_feat = feat @ Ws.T ; dst_feat = feat @ Wd.T + bd   (WMMA f32)
    linear32_wmma<<<gemm_grid, 256, 0, stream>>>(feat, Ws, nullptr, src_feat, N);
    linear32_wmma<<<gemm_grid, 256, 0, stream>>>(feat, Wd, bd,      dst_feat, N);

    // 3) agg = deg * dst_feat, then scatter-add src contributions
    init_agg_kernel<<<node4_grid, 256, 0, stream>>>(deg, dst_feat, agg, N * 8);
    scatter_add_kernel<<<edge_grid, 256, 0, stream>>>(src, dst, src_feat, agg, E);

    // 4) out = agg @ Wr.T + br
    linear32_wmma<<<gemm_grid, 256, 0, stream>>>(agg, Wr, br, outbuf, N);

    // 5) src2 = out @ Ws.T ; dst2 = out @ Wd.T + bd
    linear32_wmma<<<gemm_grid, 256, 0, stream>>>(outbuf, Ws, nullptr, src2, N);
    linear32_wmma<<<gemm_grid, 256, 0, stream>>>(outbuf, Wd, bd,      dst2, N);

    // 6) out_feat[e] = src2[src[e]] + dst2[dst[e]]
    edge_out_kernel<<<edge_grid, 256, 0, stream>>>(src, dst, src2, dst2, out, E);
}